// FootAndBall_79963701117680
// MI455X (gfx1250) — compile-verified
//
#include <hip/hip_runtime.h>
#include <hip/hip_bf16.h>
#include <stdint.h>

#define BLK 256

// ---- problem constants (from reference) ----
#define NB   64
#define HP_  68
#define WP_  120
#define HB_  272
#define WB_  480
#define MAXDET 100

typedef unsigned int u32x4 __attribute__((ext_vector_type(4)));
typedef int          i32x8 __attribute__((ext_vector_type(8)));
typedef int          i32x4 __attribute__((ext_vector_type(4)));

#if defined(__has_builtin)
#  if __has_builtin(__builtin_amdgcn_tensor_load_to_lds) && __has_builtin(__builtin_amdgcn_s_wait_tensorcnt)
#    define USE_TDM 1
#  endif
#endif

#ifdef USE_TDM
// Issue one 2-D TDM load: rows x w float32 elements, row stride == w (contiguous
// region), global -> LDS.  D# built per CDNA5 ISA §8.3 (group0) / §8.4 (group1).
// 6-arg builtin form (clang-23 / therock-10.0): (v4u g0, v8i g1, v4i g2, v4i g3, v8i, i32 cpol)
__device__ __forceinline__ void tdm_load_f32_2d(const float* gsrc, float* lds_dst,
                                                int w, int rows) {
  const unsigned long long ga  = (unsigned long long)(uintptr_t)gsrc;
  const unsigned           lds = (unsigned)(uintptr_t)lds_dst; // low 32 bits = LDS byte addr
  u32x4 g0;
  g0[0] = 1u;                                                   // count=1 (valid user D#)
  g0[1] = lds;                                                  // lds_addr
  g0[2] = (unsigned)ga;                                         // global_addr[31:0]
  g0[3] = (unsigned)((ga >> 32) & 0x01ffffffull) | (2u << 30);  // global_addr[56:32] | type=2
  i32x8 g1;
  g1[0] = (int)(2u << 16);                                         // data_size=2 (4 bytes)
  g1[1] = (int)(((unsigned)w & 0xffffu) << 16);                    // tensor_dim0[15:0]
  g1[2] = (int)((((unsigned)w >> 16) & 0xffffu) |
                (((unsigned)rows & 0xffffu) << 16));               // tensor_dim0[31:16] | tensor_dim1[15:0]
  g1[3] = (int)((((unsigned)rows >> 16) & 0xffffu) |
                (((unsigned)w & 0xffffu) << 16));                  // tensor_dim1[31:16] | tile_dim0
  g1[4] = (int)((unsigned)rows & 0xffffu);                         // tile_dim1 (tile_dim2=0)
  g1[5] = (int)(unsigned)w;                                        // tensor_dim0_stride[31:0]
  g1[6] = 0;                                                       // stride hi / dim1_stride lo
  g1[7] = 0;
  i32x4 z4; z4[0] = z4[1] = z4[2] = z4[3] = 0;                     // groups 2/3 unused (2-D)
  i32x8 z8; z8[0]=z8[1]=z8[2]=z8[3]=z8[4]=z8[5]=z8[6]=z8[7]=0;
  __builtin_amdgcn_tensor_load_to_lds(g0, g1, z4, z4, z8, 0);
}
#endif

// -------- Phase 1: per-chunk NMS + top-100 candidates --------
// fmap layout [B,2,H,W]; one block handles RPC rows of one batch image.
template <int H, int W, int RPC, int NCH>
__global__ void __launch_bounds__(BLK)
phase1_kernel(const float* __restrict__ fmap,
              float* __restrict__ cand_s, int* __restrict__ cand_i) {
  constexpr int NPIX = RPC * W;        // pixels owned by this chunk
  constexpr int HALO = (RPC + 2) * W;  // staged rows incl. halo
  __shared__ float sh_a[HALO];         // ch0 logits, then d = l1-l0
  __shared__ float sh_b[HALO];         // ch1 logits
  __shared__ float sh_s[NPIX];         // NMS'd scores
  __shared__ float red_s[BLK];
  __shared__ int   red_i[BLK];

  const int b   = blockIdx.x / NCH;
  const int c   = blockIdx.x % NCH;
  const int r0  = c * RPC;
  const int r0h = (r0 > 0) ? (r0 - 1) : 0;
  const int r1h = ((r0 + RPC + 1) < H) ? (r0 + RPC + 1) : H;
  const int nh  = r1h - r0h;
  const int nel = nh * W;

  const float* p0 = fmap + (((size_t)b * 2 + 0) * H + r0h) * (size_t)W;
  const float* p1 = fmap + (((size_t)b * 2 + 1) * H + r0h) * (size_t)W;

#ifdef USE_TDM
  if ((threadIdx.x >> 5) == 0) {       // wave 0 issues the DMAs (uniform branch)
    tdm_load_f32_2d(p0, sh_a, W, nh);
    tdm_load_f32_2d(p1, sh_b, W, nh);
    __builtin_amdgcn_s_wait_tensorcnt(0);
  }
#else
  for (int i = threadIdx.x; i < nel; i += BLK) { sh_a[i] = p0[i]; sh_b[i] = p1[i]; }
#endif
  __syncthreads();

  // d = l1 - l0 (softmax positive-class prob = sigmoid(d), monotone -> NMS on d)
  for (int i = threadIdx.x; i < nel; i += BLK) sh_a[i] = sh_b[i] - sh_a[i];
  __syncthreads();

  // 3x3 NMS ('SAME' with -inf padding) + sigmoid score, all reads from LDS
  for (int p = threadIdx.x; p < NPIX; p += BLK) {
    const int gy = r0 + p / W;
    const int gx = p % W;
    const float d0 = sh_a[(gy - r0h) * W + gx];
    bool keep = true;
#pragma unroll
    for (int dy = -1; dy <= 1; ++dy)
#pragma unroll
      for (int dx = -1; dx <= 1; ++dx) {
        if (dy == 0 && dx == 0) continue;
        const int ny = gy + dy, nx = gx + dx;
        if (ny >= 0 && ny < H && nx >= 0 && nx < W)
          keep = keep && (d0 >= sh_a[(ny - r0h) * W + nx]);
      }
    sh_s[p] = keep ? (1.0f / (1.0f + expf(-d0))) : 0.0f;
  }
  __syncthreads();

  // iterative top-100 (score desc, index asc) within the chunk
  const int tid = threadIdx.x;
  const int obase = (b * NCH + c) * MAXDET;
  for (int k = 0; k < MAXDET; ++k) {
    float bs = -2.0f; int bi = 0x7fffffff;
    for (int p = tid; p < NPIX; p += BLK) {
      const float s = sh_s[p];
      if (s > bs || (s == bs && p < bi)) { bs = s; bi = p; }
    }
    red_s[tid] = bs; red_i[tid] = bi;
    __syncthreads();
    for (int off = BLK >> 1; off > 0; off >>= 1) {
      if (tid < off) {
        const float os = red_s[tid + off]; const int oi = red_i[tid + off];
        if (os > red_s[tid] || (os == red_s[tid] && oi < red_i[tid])) {
          red_s[tid] = os; red_i[tid] = oi;
        }
      }
      __syncthreads();
    }
    if (tid == 0) {
      cand_s[obase + k] = red_s[0];
      cand_i[obase + k] = r0 * W + red_i[0];   // global flat index in h*w
      sh_s[red_i[0]] = -1.0f;                  // remove from further rounds
    }
    __syncthreads();
  }
}

// -------- Phase 2: merge chunk candidates, select global top-100, decode boxes --------
template <int CMAX>
__global__ void __launch_bounds__(BLK)
phase2_kernel(const float* __restrict__ cand_s, const int* __restrict__ cand_i,
              const float* __restrict__ bbox,  // [B,4,H,W] for player, nullptr for ball
              float* __restrict__ out,         // [B,100,5] for this map
              int C, int H, int W, float ds, int is_player) {
  __shared__ float shs[CMAX];
  __shared__ int   shi[CMAX];
  __shared__ float red_s[BLK];
  __shared__ int   red_g[BLK];
  __shared__ int   red_p[BLK];

  const int b = blockIdx.x;
  const int tid = threadIdx.x;
  for (int i = tid; i < C; i += BLK) {
    shs[i] = cand_s[(size_t)b * C + i];
    shi[i] = cand_i[(size_t)b * C + i];
  }
  __syncthreads();

  const float ctr = 0.5f * (ds - 1.0f);
  for (int k = 0; k < MAXDET; ++k) {
    float bs = -2.0f; int bg = 0x7fffffff; int bp = 0;
    for (int i = tid; i < C; i += BLK) {
      const float s = shs[i]; const int g = shi[i];
      if (s > bs || (s == bs && g < bg)) { bs = s; bg = g; bp = i; }
    }
    red_s[tid] = bs; red_g[tid] = bg; red_p[tid] = bp;
    __syncthreads();
    for (int off = BLK >> 1; off > 0; off >>= 1) {
      if (tid < off) {
        const float os = red_s[tid + off]; const int og = red_g[tid + off];
        if (os > red_s[tid] || (os == red_s[tid] && og < red_g[tid])) {
          red_s[tid] = os; red_g[tid] = og; red_p[tid] = red_p[tid + off];
        }
      }
      __syncthreads();
    }
    if (tid == 0) {
      const float v   = red_s[0];
      const int   idx = red_g[0];
      shs[red_p[0]] = -1.0f;
      const int iy = idx / W, ix = idx - iy * W;
      const float xc = (float)ix * ds + ctr;
      const float yc = (float)iy * ds + ctr;
      float t0 = 0.0f, t1 = 0.0f, t2, t3;
      if (is_player) {
        const size_t plane = (size_t)H * W;
        const float sx = (float)W * ds, sy = (float)H * ds;
        t0 = bbox[((size_t)b * 4 + 0) * plane + idx] * sx;
        t1 = bbox[((size_t)b * 4 + 1) * plane + idx] * sy;
        t2 = bbox[((size_t)b * 4 + 2) * plane + idx] * sx;
        t3 = bbox[((size_t)b * 4 + 3) * plane + idx] * sy;
      } else {
        t2 = 40.0f; t3 = 40.0f;   // BALL_BBOX_SIZE
      }
      const float bx = xc + t0, by = yc + t1;
      float* o = out + ((size_t)b * MAXDET + k) * 5;
      o[0] = bx - 0.5f * t2;
      o[1] = by - 0.5f * t3;
      o[2] = bx + 0.5f * t2;
      o[3] = by + 0.5f * t3;
      o[4] = v;
    }
    __syncthreads();
  }
}

extern "C" void kernel_launch(void* const* d_in, const int* in_sizes, int n_in,
                              void* d_out, int out_size, void* d_ws, size_t ws_size,
                              hipStream_t stream) {
  (void)in_sizes; (void)n_in; (void)out_size; (void)ws_size;
  const float* player_fm   = (const float*)d_in[0];  // [64,2,68,120]
  const float* player_bbox = (const float*)d_in[1];  // [64,4,68,120]
  const float* ball_fm     = (const float*)d_in[2];  // [64,2,272,480]
  float* out = (float*)d_out;                        // [64,100,5] player ++ [64,100,5] ball

  // player: 4 chunks of 17 rows; ball: 34 chunks of 8 rows
  constexpr int PNCH = 4, BNCH = 34;
  constexpr int PC = PNCH * MAXDET;   // 400 candidates / batch
  constexpr int BC = BNCH * MAXDET;   // 3400 candidates / batch

  float* cps = (float*)d_ws;                 // player cand scores [64*400]
  int*   cpi = (int*)(cps + NB * PC);        // player cand indices
  float* cbs = (float*)(cpi + NB * PC);      // ball cand scores   [64*3400]
  int*   cbi = (int*)(cbs + NB * BC);        // ball cand indices

  phase1_kernel<HP_, WP_, 17, PNCH><<<NB * PNCH, BLK, 0, stream>>>(player_fm, cps, cpi);
  phase1_kernel<HB_, WB_, 8,  BNCH><<<NB * BNCH, BLK, 0, stream>>>(ball_fm,   cbs, cbi);

  phase2_kernel<PC><<<NB, BLK, 0, stream>>>(cps, cpi, player_bbox, out,
                                            PC, HP_, WP_, 16.0f, 1);
  phase2_kernel<BC><<<NB, BLK, 0, stream>>>(cbs, cbi, nullptr, out + NB * MAXDET * 5,
                                            BC, HB_, WB_, 4.0f, 0);
}